// ConvDual_65481071397002
// MI455X (gfx1250) — compile-verified
//
#include <hip/hip_runtime.h>

typedef float v2f __attribute__((ext_vector_type(2)));
typedef float v8f __attribute__((ext_vector_type(8)));

#define IMG     128
#define PW      127
#define NPATCH  (127 * 127)

__global__ __launch_bounds__(256) void qconv_wmma_kernel(
    const float* __restrict__ x,       // (B,1,128,128)
    const float* __restrict__ conv_w,  // 4
    const float* __restrict__ conv_b,  // 1
    const float* __restrict__ ry,      // (2,4)
    const float* __restrict__ head_w,  // 2
    const float* __restrict__ head_b,  // 1
    float* __restrict__ out)           // (B,1)
{
    __shared__ float Vlds[16][16];     // V[j][col]  : circuit matrix columns
    __shared__ float Mlds[16][16];     // M~ = V^T S V with parity cross-blocks zeroed
    __shared__ float Xlds[8][16][33];  // per-wave 16(state) x 32(patch) staging (+pad)
    __shared__ float red[256];

    const int tid  = threadIdx.x;
    const int lane = tid & 31;
    const int wv   = tid >> 5;
    const int row  = lane & 15;
    const int hi   = lane >> 4;

    // ---- Stage 1: build V columns (16 tiny in-register circuit sims) ----
    if (tid < 16) {
        float v[16];
        #pragma unroll
        for (int j = 0; j < 16; ++j) v[j] = (j == tid) ? 1.f : 0.f;
        #pragma unroll
        for (int l = 0; l < 2; ++l) {
            #pragma unroll
            for (int w = 0; w < 4; ++w) {
                float sA, cA;
                sincosf(0.5f * ry[l * 4 + w], &sA, &cA);
                const int bit = 8 >> w;            // wire w -> bit (MSB-first)
                #pragma unroll
                for (int i0 = 0; i0 < 16; ++i0) {
                    if (!(i0 & bit)) {
                        const int i1 = i0 | bit;
                        float a0 = v[i0], a1 = v[i1];
                        v[i0] = cA * a0 - sA * a1;  // RY
                        v[i1] = sA * a0 + cA * a1;
                    }
                }
            }
            // CNOT(ctrl=wire0,bit8 ; tgt=wire1,bit4)
            #pragma unroll
            for (int i = 8; i < 12; ++i) { float t = v[i]; v[i] = v[i + 4]; v[i + 4] = t; }
            // CNOT(ctrl=wire2,bit2 ; tgt=wire3,bit1)
            #pragma unroll
            for (int i = 2; i < 16; i += 4) { float t = v[i]; v[i] = v[i + 1]; v[i + 1] = t; }
        }
        #pragma unroll
        for (int j = 0; j < 16; ++j) Vlds[j][tid] = v[j];
    }
    __syncthreads();

    // ---- Stage 2: M~[i][j] = sum_k sgn(k) V[k][i] V[k][j], parity-filtered ----
    {
        const int i = tid >> 4, j = tid & 15;
        float m = 0.f;
        if (((__popc(i) ^ __popc(j)) & 1) == 0) {
            #pragma unroll
            for (int k = 0; k < 16; ++k) {
                const float t = Vlds[k][i] * Vlds[k][j];
                m += (k < 8) ? t : -t;             // S = Z on wire 0 (MSB)
            }
        }
        Mlds[i][j] = m;
    }
    __syncthreads();

    // ---- Stage 3: constant A fragments (M~), ISA layout: M=lane%16, K=vgpr+2*hi ----
    v2f afrag[4];
    #pragma unroll
    for (int t = 0; t < 4; ++t) {
        afrag[t].x = Mlds[row][4 * t + 2 * hi];
        afrag[t].y = Mlds[row][4 * t + 1 + 2 * hi];
    }

    const float w00 = conv_w[0], w01 = conv_w[1], w10 = conv_w[2], w11 = conv_w[3];
    const float cb = conv_b[0], hw0 = head_w[0], hw1 = head_w[1], hb = head_b[0];
    const float* img = x + (size_t)blockIdx.x * (IMG * IMG);

    // sign(b) = Re/Im-merged phase of (-i)^popcount(b)
    const float sgn[16] = { 1.f,-1.f,-1.f,-1.f, -1.f,-1.f,-1.f, 1.f,
                           -1.f,-1.f,-1.f, 1.f, -1.f, 1.f, 1.f, 1.f };

    float qacc = 0.f, cacc = 0.f;
    const int iters = (NPATCH + 255) / 256;

    for (int it = 0; it < iters; ++it) {
        const int pidx  = it * 256 + tid;
        const float valid = (pidx < NPATCH) ? 1.f : 0.f;
        const int pc = (pidx < NPATCH) ? pidx : 0;
        const int py = pc / PW;
        const int px = pc - py * PW;
        const float* p0 = img + py * IMG + px;
        const float x00 = p0[0], x01 = p0[1], x10 = p0[IMG], x11 = p0[IMG + 1];

        cacc += valid * (x00 * w00 + x01 * w01 + x10 * w10 + x11 * w11 + cb);

        float c0, s0, c1, s1, c2, s2, c3, s3;
        sincosf(0.5f * x00, &s0, &c0);   // wire0
        sincosf(0.5f * x01, &s1, &c1);   // wire1
        sincosf(0.5f * x10, &s2, &c2);   // wire2
        sincosf(0.5f * x11, &s3, &c3);   // wire3
        const float p01[4] = { c0 * c1, c0 * s1, s0 * c1, s0 * s1 };
        const float p23[4] = { c2 * c3, c2 * s3, s2 * c3, s2 * s3 };

        #pragma unroll
        for (int b = 0; b < 16; ++b)
            Xlds[wv][b][lane] = valid * sgn[b] * (p01[b >> 2] * p23[b & 3]);
        __syncthreads();

        // two groups of 16 patches per wave; Y = M~ X via 4-step K-chain of WMMA
        #pragma unroll
        for (int g = 0; g < 2; ++g) {
            const int pcol = g * 16 + row;
            v8f cfrag = {0.f, 0.f, 0.f, 0.f, 0.f, 0.f, 0.f, 0.f};
            #pragma unroll
            for (int t = 0; t < 4; ++t) {
                v2f bfrag;
                bfrag.x = Xlds[wv][4 * t + 2 * hi][pcol];
                bfrag.y = Xlds[wv][4 * t + 1 + 2 * hi][pcol];
                cfrag = __builtin_amdgcn_wmma_f32_16x16x4_f32(
                    false, afrag[t], false, bfrag, (short)0, cfrag, false, false);
            }
            // column dot: q_p += x_p . (M~ x)_p ; lane covers rows hi*8..hi*8+7
            float d = 0.f;
            #pragma unroll
            for (int v = 0; v < 8; ++v)
                d += Xlds[wv][v + 8 * hi][pcol] * cfrag[v];
            qacc += d;
        }
        __syncthreads();
    }

    // ---- final block reduction; add per-patch bias, take mean ----
    red[tid] = hw1 * qacc + hw0 * cacc;
    __syncthreads();
    for (int s = 128; s > 0; s >>= 1) {
        if (tid < s) red[tid] += red[tid + s];
        __syncthreads();
    }
    if (tid == 0)
        out[blockIdx.x] = (red[0] + hb * (float)NPATCH) / (float)NPATCH;
}

extern "C" void kernel_launch(void* const* d_in, const int* in_sizes, int n_in,
                              void* d_out, int out_size, void* d_ws, size_t ws_size,
                              hipStream_t stream) {
    (void)n_in; (void)d_ws; (void)ws_size; (void)out_size;
    const float* x      = (const float*)d_in[0];
    const float* conv_w = (const float*)d_in[1];
    const float* conv_b = (const float*)d_in[2];
    const float* ry     = (const float*)d_in[3];
    const float* head_w = (const float*)d_in[4];
    const float* head_b = (const float*)d_in[5];
    float* out = (float*)d_out;
    const int B = in_sizes[0] / (IMG * IMG);   // 64
    qconv_wmma_kernel<<<B, 256, 0, stream>>>(x, conv_w, conv_b, ry, head_w, head_b, out);
}